// AudioVisualBlock_53463752900824
// MI455X (gfx1250) — compile-verified
//
#include <hip/hip_runtime.h>

// ---------------------------------------------------------------------------
// AudioVisual spiking cross-attention block for MI455X (gfx1250, wave32).
// All contractions use V_WMMA_F32_16X16X4_F32 (exact f32 - LIF thresholds
// binarize near 1.0, so reduced precision would flip spikes).
// Data movement: token dim padded to 224 so every GEMM global access is an
// unguarded b128; K-chunks stream global->LDS via async-to-LDS (ASYNCcnt),
// double-buffered to overlap with the matrix pipe. Steady-state loop is
// branch-free (last iteration peeled).
// ---------------------------------------------------------------------------

#define TBATCH 32      // T*B
#define CCH    256     // channels C
#define NN     196     // tokens N
#define NHD    16      // heads
#define DH     16      // head dim
#define HDIM   1024    // MLP hidden
#define NP     208     // 13*16 (attention tiling)
#define NT     13      // 16-row tiles covering N
#define NP2    224     // 7*32  (GEMM column padding; pad cols always zero)
#define EPSV   1e-5f

typedef __attribute__((ext_vector_type(2))) float v2f;
typedef __attribute__((ext_vector_type(8))) float v8f;
typedef __attribute__((ext_vector_type(4))) int   v4i;

#define AS1 __attribute__((address_space(1)))
#define AS3 __attribute__((address_space(3)))

#if defined(__AMDGCN__) && __has_builtin(__builtin_amdgcn_global_load_async_to_lds_b128)
#define HAVE_ASYNC 1
#else
#define HAVE_ASYNC 0
#endif

#if HAVE_ASYNC
#if __has_builtin(__builtin_amdgcn_s_wait_asynccnt)
#define WAIT_ASYNC(n) __builtin_amdgcn_s_wait_asynccnt(n)
#else
#define WAIT_ASYNC(n) asm volatile("s_wait_asynccnt %0" ::"n"(n) : "memory")
#endif
#else
#define WAIT_ASYNC(n) ((void)0)
#endif

__device__ __forceinline__ v8f wmma4(v2f a, v2f b, v8f c) {
  return __builtin_amdgcn_wmma_f32_16x16x4_f32(false, a, false, b,
                                               (short)0, c, false, false);
}

// ---------------------------------------------------------------------------
// GEMM: Cout[b,d,n] = sum_c A[d,c] * Bact[b,c,n] (+bias[d])
// A: [Dout x Cin] row-major; Bact/Cout: [TBATCH, *, NP2] (pad cols zero).
// 8 waves in a 4x2 layout over a 64x32 tile; K staged in 32-wide chunks,
// double-buffered through LDS with async-to-LDS copies.
// ---------------------------------------------------------------------------
__global__ __launch_bounds__(256) void gemm_f32_wmma_k(
    const float* __restrict__ A, const float* __restrict__ Bact,
    float* __restrict__ Cout, const float* __restrict__ bias,
    int Dout, int Cin) {
  __shared__ float As[2][64][36];   // 36-float rows: 16B-aligned for b128
  __shared__ float Bs[2][32][36];

  const int tid  = threadIdx.x;
  const int lane = tid & 31;            // wave32
  const int wid  = tid >> 5;
  const int wr   = wid >> 1;            // 0..3
  const int wc   = wid & 1;             // 0..1
  const int l16  = lane & 15;
  const int lhi  = lane >> 4;

  const int n0    = blockIdx.x * 32;
  const int row0  = blockIdx.y * 64;
  const int batch = blockIdx.z;
  const float* Bm = Bact + (size_t)batch * Cin * NP2;

  // Per-chunk staging: each thread moves 2x16B of A and 1x16B of B.
  const int ra = tid >> 3;              // 0..31 (and +32 for second A quad)
  const int ca = (tid & 7) * 4;         // 0..28

  auto stage = [&](int p, int kc) {
#if HAVE_ASYNC
#pragma unroll
    for (int i = 0; i < 2; ++i) {
      const float* g = A + (size_t)(row0 + ra + 32 * i) * Cin + kc + ca;
      __builtin_amdgcn_global_load_async_to_lds_b128(
          (AS1 v4i*)g, (AS3 v4i*)&As[p][ra + 32 * i][ca], 0, 0);
    }
    {
      const float* g = Bm + (size_t)(kc + ra) * NP2 + n0 + ca;
      __builtin_amdgcn_global_load_async_to_lds_b128(
          (AS1 v4i*)g, (AS3 v4i*)&Bs[p][ra][ca], 0, 0);
    }
#else
#pragma unroll
    for (int i = 0; i < 2; ++i) {
      const float4 va =
          *(const float4*)(A + (size_t)(row0 + ra + 32 * i) * Cin + kc + ca);
      *(float4*)&As[p][ra + 32 * i][ca] = va;
    }
    const float4 vb = *(const float4*)(Bm + (size_t)(kc + ra) * NP2 + n0 + ca);
    *(float4*)&Bs[p][ra][ca] = vb;
#endif
  };

  v8f acc = {};
  auto mma = [&](int p) {
#pragma unroll
    for (int ks = 0; ks < 8; ++ks) {
      const int k0 = ks * 4 + 2 * lhi;
      const int mr = wr * 16 + l16;
      const int nc = wc * 16 + l16;
      v2f a, b;
      a.x = As[p][mr][k0];      a.y = As[p][mr][k0 + 1];
      b.x = Bs[p][k0][nc];      b.y = Bs[p][k0 + 1][nc];
      acc = wmma4(a, b, acc);
    }
  };

  stage(0, 0);
  int p = 0;
  // Steady state: branch-free body (prefetch next, retire current, compute).
  for (int kc = 0; kc + 32 < Cin; kc += 32, p ^= 1) {
    stage(p ^ 1, kc + 32);
    WAIT_ASYNC(3);
    __syncthreads();
    mma(p);
    __syncthreads();                    // reads done before buffer reuse
  }
  // Peeled last chunk.
  WAIT_ASYNC(0);
  __syncthreads();
  mma(p);

#pragma unroll
  for (int r = 0; r < 8; ++r) {
    const int grow = row0 + wr * 16 + r + 8 * lhi;  // M = r + 8*(lane>=16)
    const int col  = n0 + wc * 16 + l16;
    float v = acc[r];
    if (bias) v += bias[grow];
    Cout[((size_t)batch * Dout + grow) * NP2 + col] = v;  // pad cols get 0(+bias)
  }
}

// ---------------------------------------------------------------------------
// Per-channel BN statistics -> fused scale/shift for the LIF apply pass.
// ---------------------------------------------------------------------------
__global__ __launch_bounds__(256) void bn_stats_k(
    const float* __restrict__ raw, const float* __restrict__ gamma,
    const float* __restrict__ beta, int Dout,
    float* __restrict__ scale, float* __restrict__ shift) {
  __shared__ float ssum[256], ssq[256];
  const int d = blockIdx.x;
  float s = 0.f, q = 0.f;
  const int total = TBATCH * NN;
  for (int i = threadIdx.x; i < total; i += 256) {
    int m = i / NN, n = i - m * NN;
    float v = raw[((size_t)m * Dout + d) * NP2 + n];
    s += v; q += v * v;
  }
  ssum[threadIdx.x] = s; ssq[threadIdx.x] = q;
  __syncthreads();
  for (int off = 128; off > 0; off >>= 1) {
    if (threadIdx.x < off) {
      ssum[threadIdx.x] += ssum[threadIdx.x + off];
      ssq[threadIdx.x]  += ssq[threadIdx.x + off];
    }
    __syncthreads();
  }
  if (threadIdx.x == 0) {
    float mu  = ssum[0] / total;
    float var = ssq[0] / total - mu * mu;
    float rs  = rsqrtf(var + EPSV);
    float g   = gamma[d];
    scale[d] = g * rs;
    shift[d] = beta[d] - g * mu * rs;
  }
}

// In-place BN+LIF; pad columns are forced to zero (keeps GEMM B-pads inert).
__global__ void bn_lif_k(float* __restrict__ z, const float* __restrict__ scale,
                         const float* __restrict__ shift, int Dout, long n4) {
  long i = (long)blockIdx.x * 256 + threadIdx.x;
  if (i >= n4) return;
  long e = i * 4;
  int n = (int)(e % NP2);
  int d = (int)((e / NP2) % Dout);
  float4 o;
  if (n < NN) {
    float4 v = *(const float4*)(z + e);
    float s = scale[d], t = shift[d];
    o.x = (v.x * s + t >= 1.f) ? 1.f : 0.f;
    o.y = (v.y * s + t >= 1.f) ? 1.f : 0.f;
    o.z = (v.z * s + t >= 1.f) ? 1.f : 0.f;
    o.w = (v.w * s + t >= 1.f) ? 1.f : 0.f;
  } else {
    o = make_float4(0.f, 0.f, 0.f, 0.f);
  }
  *(float4*)(z + e) = o;
}

// Copy unpadded [.., NN] tensor into padded [.., NP2] tensor (pads = 0).
__global__ void pack_pad_k(const float* __restrict__ src, float* __restrict__ dst,
                           long n4) {
  long i = (long)blockIdx.x * 256 + threadIdx.x;
  if (i >= n4) return;
  long e = i * 4;
  int n = (int)(e % NP2);
  long row = e / NP2;
  float4 v = (n < NN) ? *(const float4*)(src + row * NN + n)
                      : make_float4(0.f, 0.f, 0.f, 0.f);
  *(float4*)(dst + e) = v;
}

// o = a + b over padded buffers (pads stay 0 + 0 = 0).
__global__ void add_pad_k(const float* __restrict__ a, const float* __restrict__ b,
                          float* __restrict__ o, long n4) {
  long i = (long)blockIdx.x * 256 + threadIdx.x;
  if (i >= n4) return;
  long e = i * 4;
  float4 va = *(const float4*)(a + e);
  float4 vb = *(const float4*)(b + e);
  *(float4*)(o + e) = make_float4(va.x + vb.x, va.y + vb.y, va.z + vb.z, va.w + vb.w);
}

// cur = LIF(zx + fc1_b + zy + fc2_b); pads forced to zero.
__global__ void cur_lif_k(const float* __restrict__ zx, const float* __restrict__ zy,
                          const float* __restrict__ b1, const float* __restrict__ b2,
                          float* __restrict__ o, long n4) {
  long i = (long)blockIdx.x * 256 + threadIdx.x;
  if (i >= n4) return;
  long e = i * 4;
  int n = (int)(e % NP2);
  int d = (int)((e / NP2) % CCH);
  float4 r;
  if (n < NN) {
    float4 a = *(const float4*)(zx + e);
    float4 b = *(const float4*)(zy + e);
    float t = b1[d] + b2[d];
    r.x = (a.x + b.x + t >= 1.f) ? 1.f : 0.f;
    r.y = (a.y + b.y + t >= 1.f) ? 1.f : 0.f;
    r.z = (a.z + b.z + t >= 1.f) ? 1.f : 0.f;
    r.w = (a.w + b.w + t >= 1.f) ? 1.f : 0.f;
  } else {
    r = make_float4(0.f, 0.f, 0.f, 0.f);
  }
  *(float4*)(o + e) = r;
}

// out (unpadded) = cur + o (both padded).
__global__ void final_out_k(const float* __restrict__ cur, const float* __restrict__ o,
                            float* __restrict__ out, long n4) {
  long i = (long)blockIdx.x * 256 + threadIdx.x;
  if (i >= n4) return;
  long e = i * 4;
  int n = (int)(e % NN);
  long row = e / NN;
  long pe = row * NP2 + n;
  float4 a = *(const float4*)(cur + pe);
  float4 b = *(const float4*)(o + pe);
  *(float4*)(out + e) = make_float4(a.x + b.x, a.y + b.y, a.z + b.z, a.w + b.w);
}

// rpb[n,m] = P[(i1-j1+14)*29 + (i2-j2+14)], n=(i1,i2), m=(j1,j2), Hs=14, M2=29
__global__ void rpb_build_k(const float* __restrict__ P, float* __restrict__ rpb) {
  int i = blockIdx.x * 256 + threadIdx.x;
  if (i >= NN * NN) return;
  int n = i / NN, m = i - n * NN;
  int i1 = n / 14, i2 = n - i1 * 14;
  int j1 = m / 14, j2 = m - j1 * 14;
  rpb[i] = P[(i1 - j1 + 14) * 29 + (i2 - j2 + 14)];
}

// ---------------------------------------------------------------------------
// Fused attention per (tb, head, 16-row tile):
// Phase 1: attn row-block (16 x 208) = q_tile * k^T (+rpb+LIF on va branch)
// Phase 2: o_tile = attn * v, K strided over the 8 waves, LDS cross-wave
//          reduce, *0.25, LIF(0.5), stored back as [tb, c, n] with zero pads.
// ---------------------------------------------------------------------------
__global__ __launch_bounds__(256) void attn_fused_k(
    const float* __restrict__ q, const float* __restrict__ k,
    const float* __restrict__ v, const float* __restrict__ rpb,
    float* __restrict__ outsp) {
  __shared__ float k_s[NP][DH];
  __shared__ float v_s[NP][DH];
  __shared__ float q_s[16][DH];
  __shared__ float attn_s[16][NP];
  __shared__ float part_s[8][16][16];

  const int tid  = threadIdx.x;
  const int lane = tid & 31;
  const int w    = tid >> 5;
  const int l16  = lane & 15;
  const int lhi  = lane >> 4;
  const int ntile = blockIdx.x;
  const int h     = blockIdx.y;
  const int tb    = blockIdx.z;
  const size_t headBase = ((size_t)tb * CCH + h * DH) * NP2;  // + d*NP2 + n

  // Spike buffers have zeroed pads, so all loads are unguarded.
  for (int idx = tid; idx < NP * DH; idx += 256) {
    int d = idx / NP, m = idx - d * NP;
    k_s[m][d] = k[headBase + (size_t)d * NP2 + m];
    v_s[m][d] = v[headBase + (size_t)d * NP2 + m];
  }
  {
    int r = tid >> 4, d = tid & 15;
    q_s[r][d] = q[headBase + (size_t)d * NP2 + (ntile * 16 + r)];
  }
  __syncthreads();

  // Phase 1 (per-wave uniform control flow; EXEC all-ones for WMMA).
  for (int j = w; j < NT; j += 8) {
    v8f acc = {};
#pragma unroll
    for (int ks = 0; ks < 4; ++ks) {            // K = D = 16
      const int k0 = ks * 4 + 2 * lhi;
      v2f a, b;
      a.x = q_s[l16][k0];           a.y = q_s[l16][k0 + 1];
      const int mc = j * 16 + l16;
      b.x = k_s[mc][k0];            b.y = k_s[mc][k0 + 1];   // B = k^T
      acc = wmma4(a, b, acc);
    }
#pragma unroll
    for (int r = 0; r < 8; ++r) {
      const int nl = r + 8 * lhi;
      const int m  = j * 16 + l16;
      float val = acc[r];
      if (rpb) {                                // va branch: LIF(attn + rpb)
        const int nabs = ntile * 16 + nl;
        val = (nabs < NN && m < NN)
                  ? ((val + rpb[(size_t)nabs * NN + m] >= 1.f) ? 1.f : 0.f)
                  : 0.f;
      }
      attn_s[nl][m] = val;
    }
  }
  __syncthreads();

  // Phase 2: K = 208 -> 52 WMMA steps strided over 8 waves.
  v8f acc = {};
  for (int ks = w; ks < NP / 4; ks += 8) {
    const int k0 = ks * 4 + 2 * lhi;
    v2f a, b;
    a.x = attn_s[l16][k0];  a.y = attn_s[l16][k0 + 1];
    b.x = v_s[k0][l16];     b.y = v_s[k0 + 1][l16];
    acc = wmma4(a, b, acc);
  }
#pragma unroll
  for (int r = 0; r < 8; ++r)
    part_s[w][r + 8 * lhi][l16] = acc[r];
  __syncthreads();

  {
    const int row = tid >> 4, col = tid & 15;
    float s = 0.f;
#pragma unroll
    for (int ww = 0; ww < 8; ++ww) s += part_s[ww][row][col];
    s *= 0.25f;
    const int nabs = ntile * 16 + row;                   // <= 207
    const float spike = (nabs < NN && s >= 0.5f) ? 1.f : 0.f;
    outsp[headBase + (size_t)col * NP2 + nabs] = spike;  // c = h*16+col
    if (ntile == NT - 1) {                               // zero cols 208..223
      const int nz = NP + (tid & 15);
      outsp[headBase + (size_t)(tid >> 4) * NP2 + nz] = 0.f;
    }
  }
}

// ---------------------------------------------------------------------------
// Host orchestration
// ---------------------------------------------------------------------------
extern "C" void kernel_launch(void* const* d_in, const int* in_sizes, int n_in,
                              void* d_out, int out_size, void* d_ws, size_t ws_size,
                              hipStream_t stream) {
  const float* x     = (const float*)d_in[0];
  const float* y     = (const float*)d_in[1];
  const float* av_w  = (const float*)d_in[2];
  const float* av_g  = (const float*)d_in[3];
  const float* av_b  = (const float*)d_in[4];
  const float* va_w  = (const float*)d_in[5];
  const float* va_g  = (const float*)d_in[6];
  const float* va_b  = (const float*)d_in[7];
  const float* P_rpb = (const float*)d_in[8];
  const float* fc1_w = (const float*)d_in[9];
  const float* fc1_b = (const float*)d_in[10];
  const float* fc2_w = (const float*)d_in[11];
  const float* fc2_b = (const float*)d_in[12];
  const float* m1_w  = (const float*)d_in[13];
  const float* m1_b  = (const float*)d_in[14];
  const float* m1_g  = (const float*)d_in[15];
  const float* m1_bb = (const float*)d_in[16];
  const float* m2_w  = (const float*)d_in[17];
  const float* m2_b  = (const float*)d_in[18];
  const float* m2_g  = (const float*)d_in[19];
  const float* m2_bb = (const float*)d_in[20];

  const size_t SZ   = (size_t)TBATCH * CCH * NN;     // unpadded out elems
  const size_t SZP  = (size_t)TBATCH * CCH * NP2;    // padded C buffer
  const size_t SZHP = (size_t)TBATCH * HDIM * NP2;   // padded Hd buffer
  float* P0 = (float*)d_ws;      // xpad -> later y1
  float* P1 = P0 + SZP;          // ypad -> later cur
  float* W0 = P1 + SZP;          // q raw/spikes -> sca out -> o raw/spikes
  float* W1 = W0 + SZP;          // k raw/spikes -> zx
  float* W2 = W1 + SZP;          // v raw/spikes -> zy
  float* W3 = W2 + SZP;          // attention output spikes
  float* X1 = W3 + SZP;          // x1
  float* H0 = X1 + SZP;          // MLP hidden (SZHP)
  float* rpbBuf = H0 + SZHP;     // 196*196
  float* scb = rpbBuf + (size_t)NN * NN;
  float* shb = scb + HDIM;
  // Total ~81 MB of workspace.

  const long SZP4 = (long)(SZP / 4), SZ4 = (long)(SZ / 4);
  auto blk = [](long n) { return (unsigned)((n + 255) / 256); };

  auto gemm = [&](const float* A, const float* Bact, float* Cout,
                  const float* bias, int Dout, int Cin) {
    dim3 g(NP2 / 32, Dout / 64, TBATCH);
    gemm_f32_wmma_k<<<g, 256, 0, stream>>>(A, Bact, Cout, bias, Dout, Cin);
  };
  auto bnlif = [&](float* raw, const float* g_, const float* b_, int Dout) {
    bn_stats_k<<<Dout, 256, 0, stream>>>(raw, g_, b_, Dout, scb, shb);
    long n4 = (long)TBATCH * Dout * NP2 / 4;
    bn_lif_k<<<blk(n4), 256, 0, stream>>>(raw, scb, shb, Dout, n4);
  };
  // Spiking cross attention; result lands in W0 (pads zero).
  auto sca = [&](const float* src_q, const float* src_kv, const float* W,
                 const float* G, const float* Bt, const float* rpbM) {
    gemm(W + 0 * CCH * CCH, src_q,  W0, nullptr, CCH, CCH);
    gemm(W + 1 * CCH * CCH, src_kv, W1, nullptr, CCH, CCH);
    gemm(W + 2 * CCH * CCH, src_kv, W2, nullptr, CCH, CCH);
    bnlif(W0, G + 0 * CCH, Bt + 0 * CCH, CCH);
    bnlif(W1, G + 1 * CCH, Bt + 1 * CCH, CCH);
    bnlif(W2, G + 2 * CCH, Bt + 2 * CCH, CCH);
    attn_fused_k<<<dim3(NT, NHD, TBATCH), 256, 0, stream>>>(W0, W1, W2, rpbM, W3);
    gemm(W + 3 * CCH * CCH, W3, W0, nullptr, CCH, CCH);
    bnlif(W0, G + 3 * CCH, Bt + 3 * CCH, CCH);
  };

  // Pack inputs into padded layout (pads zero).
  pack_pad_k<<<blk(SZP4), 256, 0, stream>>>(x, P0, SZP4);
  pack_pad_k<<<blk(SZP4), 256, 0, stream>>>(y, P1, SZP4);

  // x1 = x + sca_av(x, y)
  sca(P0, P1, av_w, av_g, av_b, nullptr);
  add_pad_k<<<blk(SZP4), 256, 0, stream>>>(P0, W0, X1, SZP4);

  // y1 = y + sca_va(y, x, rpb)   (y1 overwrites xpad, which is now dead)
  rpb_build_k<<<(NN * NN + 255) / 256, 256, 0, stream>>>(P_rpb, rpbBuf);
  sca(P1, P0, va_w, va_g, va_b, rpbBuf);
  add_pad_k<<<blk(SZP4), 256, 0, stream>>>(P1, W0, P0, SZP4);

  // cur = LIF(fc1(x1) + fc1_b + fc2(y1) + fc2_b)   (cur overwrites ypad)
  gemm(fc1_w, X1, W1, nullptr, CCH, CCH);
  gemm(fc2_w, P0, W2, nullptr, CCH, CCH);
  cur_lif_k<<<blk(SZP4), 256, 0, stream>>>(W1, W2, fc1_b, fc2_b, P1, SZP4);

  // MLP: h = LIF(bn(m1 cur + b)); o = LIF(bn(m2 h + b)); out = cur + o
  gemm(m1_w, P1, H0, m1_b, HDIM, CCH);
  bnlif(H0, m1_g, m1_bb, HDIM);
  gemm(m2_w, H0, W0, m2_b, CCH, HDIM);
  bnlif(W0, m2_g, m2_bb, CCH);
  final_out_k<<<blk(SZ4), 256, 0, stream>>>(P1, W0, (float*)d_out, SZ4);
}